// RelationEmbeddingUpdater_36636071035733
// MI455X (gfx1250) — compile-verified
//
#include <hip/hip_runtime.h>

typedef __attribute__((ext_vector_type(2))) float v2f;
typedef __attribute__((ext_vector_type(8))) float v8f;

#define D_DIM    256
#define ROWS     128          // rows staged per block
#define RSTRIDE  260          // padded LDS row stride (floats) -> conflict-free A reads
#define NTHREADS 512          // 16 wave32s; one 16-col tile per wave

__global__ __launch_bounds__(NTHREADS, 1)
void relemb_wmma_f32_kernel(const float* __restrict__ emb,
                            const long long* __restrict__ ntype,
                            const float* __restrict__ W,
                            const float* __restrict__ bias,
                            float* __restrict__ out,
                            int n)
{
    extern __shared__ float lds[];
    float* sEmb  = lds;                          // ROWS * RSTRIDE floats
    int*   sType = (int*)(lds + ROWS * RSTRIDE); // ROWS ints

    const int tid = threadIdx.x;
    const long long rowBase = (long long)blockIdx.x * ROWS;

    // ---- Stage 128x256 fp32 emb tile via async global->LDS DMA (no VGPR bounce) ----
    // 128 rows * 64 float4 = 8192 b128 transfers; 512 threads -> 16 each.
    #pragma unroll
    for (int j = 0; j < 16; ++j) {
        int idx4 = j * NTHREADS + tid;
        int r  = idx4 >> 6;            // row in tile
        int c4 = (idx4 & 63) << 2;     // column (float index)
        long long gr = rowBase + r;
        if (gr >= n) gr = n - 1;       // clamp (stores are guarded later)
        const float* gp = emb + gr * D_DIM + c4;
        // Flat LDS pointer low 32 bits == LDS byte address (aperture mapping).
        unsigned lofs = (unsigned)(size_t)(sEmb + r * RSTRIDE + c4);
        asm volatile("global_load_async_to_lds_b128 %0, %1, off"
                     :: "v"(lofs), "v"(gp) : "memory");
    }
    if (tid < ROWS) {
        long long gr = rowBase + tid;
        if (gr >= n) gr = n - 1;
        sType[tid] = (int)ntype[gr];   // node_type is int64 in the reference
    }
    asm volatile("s_wait_asynccnt 0" ::: "memory");
    __syncthreads();

    const int wave = tid >> 5;         // 0..15 -> column tile
    const int lane = tid & 31;
    const int lm   = lane & 15;        // M (A) / N (B,C,D) within 16x16 tile
    const int lh   = lane >> 4;        // selects K pair {0,1} vs {2,3}
    const int cj   = wave << 4;        // column base of this wave's tile

    // B fragment: B[k][nj] = W[nj][k]; lane reads W row (cj+lm), floats [k+2*lh, k+2*lh+1]
    const float* wp  = W + (long long)(cj + lm) * D_DIM + 2 * lh;
    const float* ap0 = sEmb + lm * RSTRIDE + 2 * lh;

    v8f acc[8];
    #pragma unroll
    for (int rt = 0; rt < 8; ++rt) acc[rt] = (v8f)0.0f;

    // ---- Main GEMM: K=256 in steps of 4, 8 row-tiles live, software-pipelined.
    // Prefetch next k-step's fragments into a second buffer while the current
    // 8 WMMAs issue, so ds_load latency is covered by a full WMMA pack
    // instead of one s_wait_dscnt 0 per WMMA.
    v2f afc[8];
    v2f bfc = *reinterpret_cast<const v2f*>(wp);
    #pragma unroll
    for (int rt = 0; rt < 8; ++rt)
        afc[rt] = *reinterpret_cast<const v2f*>(ap0 + rt * 16 * RSTRIDE);

    #pragma unroll 2
    for (int k = 0; k < D_DIM - 4; k += 4) {
        v2f afn[8];
        const v2f bfn = *reinterpret_cast<const v2f*>(wp + k + 4);
        #pragma unroll
        for (int rt = 0; rt < 8; ++rt)
            afn[rt] = *reinterpret_cast<const v2f*>(ap0 + (k + 4) + rt * 16 * RSTRIDE);

        #pragma unroll
        for (int rt = 0; rt < 8; ++rt)
            acc[rt] = __builtin_amdgcn_wmma_f32_16x16x4_f32(
                false, afc[rt], false, bfc, (short)0, acc[rt], false, false);

        #pragma unroll
        for (int rt = 0; rt < 8; ++rt) afc[rt] = afn[rt];
        bfc = bfn;
    }
    #pragma unroll
    for (int rt = 0; rt < 8; ++rt)
        acc[rt] = __builtin_amdgcn_wmma_f32_16x16x4_f32(
            false, afc[rt], false, bfc, (short)0, acc[rt], false, false);

    // ---- Epilogue: bias add, per-row type select, guarded coalesced stores ----
    // C/D layout: VGPR i, lane L -> (M = i + 8*(L/16), N = L%16)
    const float bv = bias[cj + lm];
    #pragma unroll
    for (int rt = 0; rt < 8; ++rt) {
        #pragma unroll
        for (int i = 0; i < 8; ++i) {
            const int lrow = rt * 16 + i + 8 * lh;   // row within block tile
            const long long grow = rowBase + lrow;
            if (grow < n) {
                const float proj = acc[rt][i] + bv;
                const float orig = sEmb[lrow * RSTRIDE + cj + lm];
                out[grow * D_DIM + cj + lm] = (sType[lrow] == 1) ? proj : orig;
            }
        }
    }
}

extern "C" void kernel_launch(void* const* d_in, const int* in_sizes, int n_in,
                              void* d_out, int out_size, void* d_ws, size_t ws_size,
                              hipStream_t stream) {
    const float*     emb   = (const float*)d_in[0];
    const long long* ntype = (const long long*)d_in[1];
    const float*     W     = (const float*)d_in[2];
    const float*     bias  = (const float*)d_in[3];
    float*           out   = (float*)d_out;

    const int n = in_sizes[1];                     // N = number of rows
    const int blocks = (n + ROWS - 1) / ROWS;
    const size_t shmem = (size_t)ROWS * RSTRIDE * sizeof(float)
                       + (size_t)ROWS * sizeof(int);

    hipLaunchKernelGGL(relemb_wmma_f32_kernel, dim3(blocks), dim3(NTHREADS),
                       shmem, stream, emb, ntype, W, bias, out, n);
}